// Net_53283364274903
// MI455X (gfx1250) — compile-verified
//
#include <hip/hip_runtime.h>
#include <hip/hip_bf16.h>
#include <cmath>

// MI455X / gfx1250 SplineConv pipeline.
// - Node features kept L2-resident: padded [N,32] fp32 (12.8 MB << 192 MB L2).
// - Edge stage: 27 spline kernel matrices staged into LDS via
//   global_load_async_to_lds_b128 (ASYNCcnt-tracked), per-edge 8-basis combine,
//   global_atomic_add_f32 scatter.
// - Dense root transform: V_WMMA_F32_16X16X4_F32, K-loop fully unrolled via
//   template FIN so fragment loads pipeline ahead of the WMMA chain.

typedef __attribute__((ext_vector_type(2))) float v2f;
typedef __attribute__((ext_vector_type(8))) float v8f;
typedef __attribute__((ext_vector_type(4))) int v4i;

#define NSTRIDE 32  // padded feature stride; padding columns kept exactly 0

#if defined(__has_builtin)
#if __has_builtin(__builtin_amdgcn_global_load_async_to_lds_b128) && \
    __has_builtin(__builtin_amdgcn_s_wait_asynccnt)
#define HAVE_ASYNC_LDS 1
#endif
#endif
#ifndef HAVE_ASYNC_LDS
#define HAVE_ASYNC_LDS 0
#endif

// ---- pack x[N,1] -> h[N,32] with zero padding ----
__global__ void pack_input(const float* __restrict__ x, float* __restrict__ h, int n) {
  int i = blockIdx.x * blockDim.x + threadIdx.x;
  if (i < n * NSTRIDE) {
    int col = i & (NSTRIDE - 1);
    int row = i >> 5;
    h[i] = (col == 0) ? x[row] : 0.0f;
  }
}

// ---- h_out[N,32] = bias + h_in @ root, via fp32 WMMA 16x16x4 ----
// grid.x = N/32 (2 waves/block, 16 rows each), grid.y = 2 (16-col tiles; both
// tiles written so padding columns come out exactly 0).
// Wave32 WMMA layouts (ISA 7.12.2):
//   A 16x4 f32 : lane l -> M=l&15, a.x=K(kb), a.y=K(kb+1), kb=2*(l>>4)
//   B 4x16 f32 : lane l -> N=l&15, b.x=row kb, b.y=row kb+1
//   C 16x16    : vgpr v, lane l -> M=v+8*(l>>4), N=l&15
template <int FIN>
__global__ void __launch_bounds__(64) root_gemm(const float* __restrict__ h_in,
                                                const float* __restrict__ root,
                                                const float* __restrict__ bias,
                                                float* __restrict__ h_out, int fout) {
  constexpr int KSTEPS = (FIN + 3) / 4;  // padded K; pad cols of h_in are 0
  __shared__ float sB[32 * 16];          // padded B tile: [k=0..31][col=0..15]
  __shared__ float sBias[16];
  const int c0 = blockIdx.y * 16;
  for (int t = threadIdx.x; t < 32 * 16; t += 64) {
    int k = t >> 4, col = t & 15, gc = c0 + col;
    sB[t] = (k < FIN && gc < fout) ? root[k * fout + gc] : 0.0f;
  }
  if (threadIdx.x < 16) {
    int gc = c0 + threadIdx.x;
    sBias[threadIdx.x] = (gc < fout) ? bias[gc] : 0.0f;
  }
  __syncthreads();

  const int lane = threadIdx.x & 31;
  const int wave = threadIdx.x >> 5;
  const int r0   = blockIdx.x * 32 + wave * 16;  // N % 32 == 0 assumed
  const int col  = lane & 15;
  const int kb   = (lane >> 4) << 1;
  const int row  = r0 + (lane & 15);

  const float bv = sBias[col];
  v8f c = {bv, bv, bv, bv, bv, bv, bv, bv};

  // Fully unrolled: all fragment loads issue before the WMMA chain.
  v2f a[KSTEPS], b[KSTEPS];
#pragma unroll
  for (int s = 0; s < KSTEPS; ++s) {
    const int k0 = s * 4;
    a[s].x = h_in[row * NSTRIDE + k0 + kb];
    a[s].y = h_in[row * NSTRIDE + k0 + kb + 1];
    b[s].x = sB[(k0 + kb) * 16 + col];
    b[s].y = sB[(k0 + kb + 1) * 16 + col];
  }
#pragma unroll
  for (int s = 0; s < KSTEPS; ++s) {
    c = __builtin_amdgcn_wmma_f32_16x16x4_f32(false, a[s], false, b[s], (short)0,
                                              c, false, false);
  }

#pragma unroll
  for (int v = 0; v < 8; ++v) {
    int m = v + ((lane >> 4) << 3);
    h_out[(r0 + m) * NSTRIDE + c0 + col] = c[v];
  }
}

// ---- per-edge spline conv, weights in LDS, atomic scatter to h_out ----
template <int FIN, int FOUT>
__global__ void __launch_bounds__(256) edge_kernel(const float* __restrict__ h_in,
                                                   const int* __restrict__ ei,
                                                   const float* __restrict__ pseudo,
                                                   const float* __restrict__ W,
                                                   float* __restrict__ h_out, int nE) {
  constexpr int ROWSZ = FIN * FOUT;     // multiple of 4 for all layers
  constexpr int ROW = ROWSZ + 4;        // +16B: alignment kept, banks staggered
  __shared__ float sW[27 * ROW];

#if HAVE_ASYNC_LDS
  {
    constexpr int CHUNKS = 27 * ROWSZ / 4;  // 16B chunks, never cross a row
    for (int cidx = threadIdx.x; cidx < CHUNKS; cidx += blockDim.x) {
      const int k  = cidx / (ROWSZ / 4);
      const int r4 = (cidx - k * (ROWSZ / 4)) * 4;
      const float* gp = W + k * ROWSZ + r4;
      float* lp = &sW[k * ROW + r4];
      __builtin_amdgcn_global_load_async_to_lds_b128(
          (__attribute__((address_space(1))) v4i*)gp,
          (__attribute__((address_space(3))) v4i*)lp, 0, 0);
    }
    __builtin_amdgcn_s_wait_asynccnt(0);
  }
#else
  for (int t = threadIdx.x; t < 27 * ROWSZ; t += blockDim.x) {
    int k = t / ROWSZ;
    int r = t - k * ROWSZ;
    sW[k * ROW + r] = W[t];
  }
#endif
  __syncthreads();

  int e = blockIdx.x * blockDim.x + threadIdx.x;
  if (e >= nE) return;
  const int src = ei[e];
  const int dst = ei[nE + e];

  float fr[3];
  int fi[3];
#pragma unroll
  for (int d = 0; d < 3; ++d) {
    float v = pseudo[e * 3 + d] * 2.0f;              // (KS-1) = 2
    float fl = fminf(fmaxf(floorf(v), 0.0f), 1.0f);  // clip to [0, KS-2]
    fr[d] = v - fl;
    fi[d] = (int)fl;
  }

  float xs[FIN];
  if constexpr (FIN % 4 == 0) {
    const float4* x4 = reinterpret_cast<const float4*>(h_in + (size_t)src * NSTRIDE);
#pragma unroll
    for (int f4 = 0; f4 < FIN / 4; ++f4) {
      float4 v = x4[f4];
      xs[f4 * 4 + 0] = v.x; xs[f4 * 4 + 1] = v.y;
      xs[f4 * 4 + 2] = v.z; xs[f4 * 4 + 3] = v.w;
    }
  } else {
#pragma unroll
    for (int f = 0; f < FIN; ++f) xs[f] = h_in[(size_t)src * NSTRIDE + f];
  }

  float acc[FOUT];
#pragma unroll
  for (int o = 0; o < FOUT; ++o) acc[o] = 0.0f;

  for (int b = 0; b < 8; ++b) {
    const int b0 = b & 1, b1 = (b >> 1) & 1, b2 = (b >> 2) & 1;
    const float w = (b0 ? fr[0] : 1.0f - fr[0]) *
                    (b1 ? fr[1] : 1.0f - fr[1]) *
                    (b2 ? fr[2] : 1.0f - fr[2]);
    const int idx = (fi[0] + b0) + (fi[1] + b1) * 3 + (fi[2] + b2) * 9;
    const float* wrow = &sW[idx * ROW];
    for (int f = 0; f < FIN; ++f) {
      const float s = w * xs[f];
      if constexpr (FOUT % 4 == 0) {
        const float4* w4 = reinterpret_cast<const float4*>(wrow + f * FOUT);
#pragma unroll
        for (int o4 = 0; o4 < FOUT / 4; ++o4) {
          float4 wv = w4[o4];
          acc[o4 * 4 + 0] += s * wv.x;
          acc[o4 * 4 + 1] += s * wv.y;
          acc[o4 * 4 + 2] += s * wv.z;
          acc[o4 * 4 + 3] += s * wv.w;
        }
      } else {
#pragma unroll
        for (int o = 0; o < FOUT; ++o) acc[o] += s * wrow[f * FOUT + o];
      }
    }
  }
#pragma unroll
  for (int o = 0; o < FOUT; ++o)
    atomicAdd(&h_out[(size_t)dst * NSTRIDE + o], acc[o]);
}

// ---- ELU in place; elu(0)=0 so zero padding is preserved ----
__global__ void elu_kernel(float* __restrict__ h, int total) {
  int i = blockIdx.x * blockDim.x + threadIdx.x;
  if (i < total) {
    float v = h[i];
    h[i] = v > 0.0f ? v : (__expf(v) - 1.0f);
  }
}

__global__ void extract_kernel(const float* __restrict__ h, float* __restrict__ out,
                               int n) {
  int i = blockIdx.x * blockDim.x + threadIdx.x;
  if (i < n) out[i] = h[(size_t)i * NSTRIDE];
}

template <int FIN, int FOUT>
static void run_layer(const float* h_in, float* h_out, const int* ei,
                      const float* pseudo, const float* W, const float* R,
                      const float* B, int N, int E, hipStream_t stream) {
  dim3 g(N / 32, 2);
  root_gemm<FIN><<<g, 64, 0, stream>>>(h_in, R, B, h_out, FOUT);
  edge_kernel<FIN, FOUT><<<(E + 255) / 256, 256, 0, stream>>>(h_in, ei, pseudo, W,
                                                              h_out, E);
  elu_kernel<<<(N * NSTRIDE + 255) / 256, 256, 0, stream>>>(h_out, N * NSTRIDE);
}

extern "C" void kernel_launch(void* const* d_in, const int* in_sizes, int n_in,
                              void* d_out, int out_size, void* d_ws, size_t ws_size,
                              hipStream_t stream) {
  const float* x      = (const float*)d_in[0];
  const int*   ei     = (const int*)d_in[1];
  const float* pseudo = (const float*)d_in[2];
  const float* W[6];
  const float* R[6];
  const float* B[6];
  for (int i = 0; i < 6; ++i) {
    W[i] = (const float*)d_in[3 + 3 * i];
    R[i] = (const float*)d_in[4 + 3 * i];
    B[i] = (const float*)d_in[5 + 3 * i];
  }
  const int N = in_sizes[0];      // x is (N,1)
  const int E = in_sizes[1] / 2;  // edge_index is (2,E)

  float* hA = (float*)d_ws;
  float* hB = hA + (size_t)N * NSTRIDE;  // needs 2*N*32*4 = 25.6 MB of ws

  pack_input<<<(N * NSTRIDE + 255) / 256, 256, 0, stream>>>(x, hA, N);
  run_layer<1, 8>(hA, hB, ei, pseudo, W[0], R[0], B[0], N, E, stream);
  run_layer<8, 16>(hB, hA, ei, pseudo, W[1], R[1], B[1], N, E, stream);
  run_layer<16, 32>(hA, hB, ei, pseudo, W[2], R[2], B[2], N, E, stream);
  run_layer<32, 16>(hB, hA, ei, pseudo, W[3], R[3], B[3], N, E, stream);
  run_layer<16, 8>(hA, hB, ei, pseudo, W[4], R[4], B[4], N, E, stream);
  run_layer<8, 1>(hB, hA, ei, pseudo, W[5], R[5], B[5], N, E, stream);
  extract_kernel<<<(N + 255) / 256, 256, 0, stream>>>(hA, (float*)d_out, N);
}